// GaussianPointCloudRasterisation_13812614824642
// MI455X (gfx1250) — compile-verified
//
#include <hip/hip_runtime.h>
#include <hip/hip_bf16.h>

typedef __attribute__((ext_vector_type(2))) float v2f;
typedef __attribute__((ext_vector_type(8))) float v8f;

#define NEARP 0.8f
#define FARP 1000.0f
#define TILEPX 16
#define SORT_T 256
#define SORT_ITEMS 16
#define SORT_CHUNK (SORT_T * SORT_ITEMS)

// ---------------------------------------------------------------------------
// Setup: invert T (Gauss-Jordan), build WMMA B operand (4x16) and constants.
// consts layout (floats): [0..63] B[k][n] row-major, [64..72] Rw row-major,
// [73] fx, [74] fy
// ---------------------------------------------------------------------------
__global__ void setup_kernel(const float* __restrict__ K,
                             const float* __restrict__ T,
                             float* __restrict__ consts) {
  if (threadIdx.x != 0 || blockIdx.x != 0) return;
  float a[4][8];
  for (int i = 0; i < 4; ++i)
    for (int j = 0; j < 4; ++j) {
      a[i][j] = T[i * 4 + j];
      a[i][j + 4] = (i == j) ? 1.f : 0.f;
    }
  for (int col = 0; col < 4; ++col) {
    int piv = col; float best = fabsf(a[col][col]);
    for (int r = col + 1; r < 4; ++r) {
      float v = fabsf(a[r][col]);
      if (v > best) { best = v; piv = r; }
    }
    if (piv != col)
      for (int j = 0; j < 8; ++j) { float t = a[col][j]; a[col][j] = a[piv][j]; a[piv][j] = t; }
    float d = 1.0f / a[col][col];
    for (int j = 0; j < 8; ++j) a[col][j] *= d;
    for (int r = 0; r < 4; ++r) if (r != col) {
      float f = a[r][col];
      for (int j = 0; j < 8; ++j) a[r][j] -= f * a[col][j];
    }
  }
  float Rw[3][3], tw[3], KR[3][3], Kt[3];
  for (int i = 0; i < 3; ++i) {
    for (int j = 0; j < 3; ++j) Rw[i][j] = a[i][4 + j];
    tw[i] = a[i][7];
  }
  for (int i = 0; i < 3; ++i) {
    for (int j = 0; j < 3; ++j) {
      float s = 0.f;
      for (int k = 0; k < 3; ++k) s += K[i * 3 + k] * Rw[k][j];
      KR[i][j] = s;
    }
    float s = 0.f;
    for (int k = 0; k < 3; ++k) s += K[i * 3 + k] * tw[k];
    Kt[i] = s;
  }
  for (int k = 0; k < 4; ++k)
    for (int n = 0; n < 16; ++n) consts[k * 16 + n] = 0.f;
  // columns 0..2: xyz_cam ; columns 3..5: proj
  for (int n = 0; n < 3; ++n) {
    for (int k = 0; k < 3; ++k) consts[k * 16 + n] = Rw[n][k];
    consts[3 * 16 + n] = tw[n];
  }
  for (int j = 0; j < 3; ++j) {
    int n = 3 + j;
    for (int k = 0; k < 3; ++k) consts[k * 16 + n] = KR[j][k];
    consts[3 * 16 + n] = Kt[j];
  }
  for (int i = 0; i < 3; ++i)
    for (int j = 0; j < 3; ++j) consts[64 + i * 3 + j] = Rw[i][j];
  consts[73] = K[0];
  consts[74] = K[4];
}

// ---------------------------------------------------------------------------
// Preprocess: WMMA fp32 16x16x4 transform+projection, then per-point mask,
// sort-key, covariance. Writes unsorted rows, int64 keys, compact 32-bit keys,
// identity index, and the (unsorted) mask output.
// ---------------------------------------------------------------------------
__global__ void preprocess_kernel(const float* __restrict__ pc,
                                  const float* __restrict__ feats,
                                  const float* __restrict__ consts,
                                  float* __restrict__ rows,
                                  long long* __restrict__ skey,
                                  unsigned* __restrict__ ckey,
                                  unsigned* __restrict__ cidx,
                                  float* __restrict__ maskOut,
                                  const int* __restrict__ camWp,
                                  const int* __restrict__ camHp,
                                  int N) {
  __shared__ float lds[8][32][8];  // [wave][point-in-wave][col], cols 0..5 used
  const int t = threadIdx.x;
  const int wave = t >> 5;
  const int lane = t & 31;
  const int half = lane >> 4;
  const int ln = lane & 15;
  const int blockBase = blockIdx.x * 256;
  const int camW = *camWp;
  const int camH = *camHp;

  const int q = blockBase + t;
  if (q < N) __builtin_prefetch(feats + (size_t)q * 56, 0, 0);

  // B operand: VGPR r holds rows K=r (lanes 0-15) and K=r+2 (lanes 16-31)
  v2f b;
  b.x = consts[(2 * half + 0) * 16 + ln];
  b.y = consts[(2 * half + 1) * 16 + ln];

  // Two 16-point WMMA tiles per wave -> 32 points per wave, 256 per block
#pragma unroll
  for (int tile = 0; tile < 2; ++tile) {
    int p = blockBase + wave * 32 + tile * 16 + ln;
    int pi = (p < N) ? p : 0;  // clamp so EXEC stays all-ones for WMMA
    // A 16x4: lanes 0-15 hold K=0,1 ; lanes 16-31 hold K=2,3
    v2f a;
    a.x = pc[(size_t)pi * 3 + (half ? 2 : 0)];
    a.y = half ? 1.0f : pc[(size_t)pi * 3 + 1];
    v8f c = {};
    v8f d = __builtin_amdgcn_wmma_f32_16x16x4_f32(false, a, false, b,
                                                  (short)0, c, false, false);
    // D: vgpr r, lanes 0-15 -> (M=r, N=lane); lanes 16-31 -> (M=r+8, N=lane-16)
#pragma unroll
    for (int r = 0; r < 8; ++r) {
      int M = r + 8 * half;
      if (ln < 6) lds[wave][tile * 16 + M][ln] = d[r];
    }
  }
  __syncthreads();

  if (q < N) {
    float xc = lds[wave][lane][0];
    float yc = lds[wave][lane][1];
    float zc = lds[wave][lane][2];
    float pu = lds[wave][lane][3];
    float pv = lds[wave][lane][4];
    float u = pu / zc;
    float v = pv / zc;
    bool mask = (zc > NEARP) && (zc < FARP) && (u >= 0.f) &&
                (u < (float)camW) && (v >= 0.f) && (v < (float)camH);
    int depth = (int)(zc * 100.0f);
    int tu = (int)(u * (1.0f / 16.0f));
    int tv = (int)(v * (1.0f / 16.0f));
    long long tid64 = (long long)(tu + tv * (camW / TILEPX));
    long long key = (long long)depth + (tid64 << 32);
    if (!mask) key = 0x7FFFFFFFFFFFFFFFLL;
    // monotone 30-bit surrogate: valid depth < 2^17, tile_id < 2^13
    unsigned ck = mask ? ((((unsigned)tid64) << 17) | (unsigned)depth)
                       : 0xFFFFFFFFu;
    skey[q] = key;
    ckey[q] = ck;
    cidx[q] = (unsigned)q;
    maskOut[q] = mask ? 1.0f : 0.0f;

    // --- covariance ---
    const float* f = feats + (size_t)q * 56;
    float q0 = f[0], q1 = f[1], q2 = f[2], q3 = f[3];
    float inv = 1.0f / sqrtf(q0 * q0 + q1 * q1 + q2 * q2 + q3 * q3);
    float x = q0 * inv, y = q1 * inv, z = q2 * inv, w = q3 * inv;
    float R00 = 1 - 2 * (y * y + z * z), R01 = 2 * (x * y - w * z), R02 = 2 * (x * z + w * y);
    float R10 = 2 * (x * y + w * z), R11 = 1 - 2 * (x * x + z * z), R12 = 2 * (y * z - w * x);
    float R20 = 2 * (x * z - w * y), R21 = 2 * (y * z + w * x), R22 = 1 - 2 * (x * x + y * y);
    float s0 = expf(f[4]), s1 = expf(f[5]), s2 = expf(f[6]);
    float M00 = R00 * s0, M01 = R01 * s1, M02 = R02 * s2;
    float M10 = R10 * s0, M11 = R11 * s1, M12 = R12 * s2;
    float M20 = R20 * s0, M21 = R21 * s1, M22 = R22 * s2;
    float C00 = M00 * M00 + M01 * M01 + M02 * M02;
    float C01 = M00 * M10 + M01 * M11 + M02 * M12;
    float C02 = M00 * M20 + M01 * M21 + M02 * M22;
    float C11 = M10 * M10 + M11 * M11 + M12 * M12;
    float C12 = M10 * M20 + M11 * M21 + M12 * M22;
    float C22 = M20 * M20 + M21 * M21 + M22 * M22;
    float fx = consts[73], fy = consts[74];
    float Rw00 = consts[64], Rw01 = consts[65], Rw02 = consts[66];
    float Rw10 = consts[67], Rw11 = consts[68], Rw12 = consts[69];
    float Rw20 = consts[70], Rw21 = consts[71], Rw22 = consts[72];
    float invz = 1.0f / zc;
    float J00 = fx * invz, J02 = -fx * xc * invz * invz;
    float J11 = fy * invz, J12 = -fy * yc * invz * invz;
    float A0 = J00 * Rw00 + J02 * Rw20;
    float A1 = J00 * Rw01 + J02 * Rw21;
    float A2 = J00 * Rw02 + J02 * Rw22;
    float B0 = J11 * Rw10 + J12 * Rw20;
    float B1 = J11 * Rw11 + J12 * Rw21;
    float B2 = J11 * Rw12 + J12 * Rw22;
    float CA0 = C00 * A0 + C01 * A1 + C02 * A2;
    float CA1 = C01 * A0 + C11 * A1 + C12 * A2;
    float CA2 = C02 * A0 + C12 * A1 + C22 * A2;
    float CB0 = C00 * B0 + C01 * B1 + C02 * B2;
    float CB1 = C01 * B0 + C11 * B1 + C12 * B2;
    float CB2 = C02 * B0 + C12 * B1 + C22 * B2;
    float cov00 = A0 * CA0 + A1 * CA1 + A2 * CA2;
    float cov01 = B0 * CA0 + B1 * CA1 + B2 * CA2;
    float cov10 = A0 * CB0 + A1 * CB1 + A2 * CB2;
    float cov11 = B0 * CB0 + B1 * CB1 + B2 * CB2;
    float mf = mask ? 1.f : 0.f;
    float* o = rows + (size_t)q * 9;
    o[0] = u * mf;  o[1] = v * mf;
    o[2] = xc * mf; o[3] = yc * mf; o[4] = zc * mf;
    o[5] = cov00 * mf; o[6] = cov01 * mf; o[7] = cov10 * mf; o[8] = cov11 * mf;
  }
}

// ---------------------------------------------------------------------------
// Stable LSD radix sort: 8 passes x 4 bits on 32-bit compact keys + u32 index
// ---------------------------------------------------------------------------
__global__ void hist_kernel(const unsigned* __restrict__ keys,
                            unsigned* __restrict__ hist,
                            int n, int shift, int numBlocks) {
  __shared__ unsigned h[16];
  int t = threadIdx.x, g = blockIdx.x;
  if (t < 16) h[t] = 0;
  __syncthreads();
  int base = g * SORT_CHUNK + t * SORT_ITEMS;
#pragma unroll
  for (int i = 0; i < SORT_ITEMS; ++i) {
    int idx = base + i;
    if (idx < n) atomicAdd(&h[(keys[idx] >> shift) & 15u], 1u);
  }
  __syncthreads();
  if (t < 16) hist[t * numBlocks + g] = h[t];
}

__global__ void scan_kernel(unsigned* __restrict__ data, int n) {
  __shared__ unsigned tmp[SORT_T];
  __shared__ unsigned carry;
  int t = threadIdx.x;
  if (t == 0) carry = 0;
  __syncthreads();
  for (int base = 0; base < n; base += SORT_T) {
    unsigned v = (base + t < n) ? data[base + t] : 0u;
    tmp[t] = v;
    __syncthreads();
    for (int off = 1; off < SORT_T; off <<= 1) {
      unsigned x = (t >= off) ? tmp[t - off] : 0u;
      __syncthreads();
      tmp[t] += x;
      __syncthreads();
    }
    unsigned incl = tmp[t];
    unsigned excl = incl - v + carry;
    if (base + t < n) data[base + t] = excl;
    __syncthreads();
    if (t == SORT_T - 1) carry += incl;
    __syncthreads();
  }
}

__global__ void scatter_kernel(const unsigned* __restrict__ keysIn,
                               const unsigned* __restrict__ idxIn,
                               unsigned* __restrict__ keysOut,
                               unsigned* __restrict__ idxOut,
                               const unsigned* __restrict__ hist,
                               int n, int shift, int numBlocks) {
  __shared__ unsigned cnt[16 * SORT_T];  // [bin*256 + thread]
  __shared__ unsigned binBase[16];
  int t = threadIdx.x, g = blockIdx.x;
#pragma unroll
  for (int b = 0; b < 16; ++b) cnt[b * SORT_T + t] = 0;
  if (t < 16) binBase[t] = hist[t * numBlocks + g];
  __syncthreads();
  int base = g * SORT_CHUNK + t * SORT_ITEMS;
  unsigned k[SORT_ITEMS], id[SORT_ITEMS];
#pragma unroll
  for (int i = 0; i < SORT_ITEMS; ++i) {
    int idx = base + i;
    if (idx < n) {
      k[i] = keysIn[idx];
      id[i] = idxIn[idx];
      cnt[((k[i] >> shift) & 15u) * SORT_T + t]++;
    }
  }
  __syncthreads();
  // exclusive scan over threads, per bin (threads 0..15 own one bin each)
  if (t < 16) {
    unsigned run = 0;
    for (int j = 0; j < SORT_T; ++j) {
      unsigned c = cnt[t * SORT_T + j];
      cnt[t * SORT_T + j] = binBase[t] + run;
      run += c;
    }
  }
  __syncthreads();
#pragma unroll
  for (int i = 0; i < SORT_ITEMS; ++i) {
    int idx = base + i;
    if (idx < n) {
      unsigned d = (k[i] >> shift) & 15u;
      unsigned pos = cnt[d * SORT_T + t]++;  // thread-private column: no race
      keysOut[pos] = k[i];
      idxOut[pos] = id[i];
    }
  }
}

__global__ void gather_kernel(const unsigned* __restrict__ perm,
                              const float* __restrict__ rows,
                              const long long* __restrict__ skey,
                              float* __restrict__ out, int N) {
  int i = blockIdx.x * blockDim.x + threadIdx.x;
  if (i < N) {
    unsigned p = perm[i];
#pragma unroll
    for (int k = 0; k < 9; ++k)
      out[(size_t)i * 9 + k] = rows[(size_t)p * 9 + k];
    out[(size_t)N * 9 + i] = (float)skey[p];
  }
}

// ---------------------------------------------------------------------------
extern "C" void kernel_launch(void* const* d_in, const int* in_sizes, int n_in,
                              void* d_out, int out_size, void* d_ws, size_t ws_size,
                              hipStream_t stream) {
  const float* pc    = (const float*)d_in[0];
  const float* feats = (const float*)d_in[1];
  const float* K     = (const float*)d_in[2];
  const float* T     = (const float*)d_in[3];
  const int* camW    = (const int*)d_in[4];
  const int* camH    = (const int*)d_in[5];
  const int N = in_sizes[0] / 3;
  const int numBlocks = (N + SORT_CHUNK - 1) / SORT_CHUNK;

  char* ws = (char*)d_ws;
  size_t off = 0;
  auto alloc = [&](size_t bytes, size_t align) -> size_t {
    off = (off + align - 1) & ~(align - 1);
    size_t r = off;
    off += bytes;
    return r;
  };
  float*     consts = (float*)(ws + alloc(128 * 4, 256));
  float*     rows   = (float*)(ws + alloc((size_t)N * 9 * 4, 256));
  long long* skey   = (long long*)(ws + alloc((size_t)N * 8, 256));
  unsigned*  keyA   = (unsigned*)(ws + alloc((size_t)N * 4, 256));
  unsigned*  keyB   = (unsigned*)(ws + alloc((size_t)N * 4, 256));
  unsigned*  idxA   = (unsigned*)(ws + alloc((size_t)N * 4, 256));
  unsigned*  idxB   = (unsigned*)(ws + alloc((size_t)N * 4, 256));
  unsigned*  hist   = (unsigned*)(ws + alloc((size_t)16 * numBlocks * 4, 256));

  float* out = (float*)d_out;
  float* maskOut = out + (size_t)10 * N;  // [0,9N) sorted rows, [9N,10N) keys, [10N,11N) mask

  setup_kernel<<<1, 1, 0, stream>>>(K, T, consts);
  preprocess_kernel<<<(N + 255) / 256, 256, 0, stream>>>(
      pc, feats, consts, rows, skey, keyA, idxA, maskOut, camW, camH, N);

  unsigned *kin = keyA, *iin = idxA, *kout = keyB, *iout = idxB;
  for (int pass = 0; pass < 8; ++pass) {
    int shift = pass * 4;
    hist_kernel<<<numBlocks, SORT_T, 0, stream>>>(kin, hist, N, shift, numBlocks);
    scan_kernel<<<1, SORT_T, 0, stream>>>(hist, 16 * numBlocks);
    scatter_kernel<<<numBlocks, SORT_T, 0, stream>>>(kin, iin, kout, iout, hist,
                                                     N, shift, numBlocks);
    unsigned* tk = kin; kin = kout; kout = tk;
    unsigned* ti = iin; iin = iout; iout = ti;
  }
  gather_kernel<<<(N + 255) / 256, 256, 0, stream>>>(iin, rows, skey, out, N);
}